// MoERouter_88330297410165
// MI455X (gfx1250) — compile-verified
//
#include <hip/hip_runtime.h>
#include <hip/hip_bf16.h>

typedef __attribute__((ext_vector_type(2))) float v2f;
typedef __attribute__((ext_vector_type(8))) float v8f;

#define T_TOK   16384   // B*N = 4*4096
#define C_DIM   2048
#define E_EXP   64
#define B_BATCH 4

// ---------------------------------------------------------------------------
// Kernel 1: ctx[b][j] = sum_k rc[b][k] * ctx_w[j][k]   (tiny 4x2048x2048 GEMV)
// ---------------------------------------------------------------------------
__global__ __launch_bounds__(256) void ctx_proj_kernel(
    const float* __restrict__ rc,      // (B, C)
    const float* __restrict__ ctx_w,   // (C, C) torch layout (out,in)
    float* __restrict__ ctx_out)       // (B, C)
{
    int t = blockIdx.x * 256 + threadIdx.x;       // 0..8191
    int b = t >> 11;                              // /2048
    int j = t & (C_DIM - 1);
    const float* a = rc + (size_t)b * C_DIM;
    const float* w = ctx_w + (size_t)j * C_DIM;
    float acc = 0.0f;
#pragma unroll 4
    for (int k = 0; k < C_DIM; k += 4) {
        float4 av = *(const float4*)(a + k);
        float4 wv = *(const float4*)(w + k);
        acc += av.x * wv.x + av.y * wv.y + av.z * wv.z + av.w * wv.w;
    }
    ctx_out[t] = acc;
}

// ---------------------------------------------------------------------------
// Kernel 2: fused logits (WMMA f32 16x16x4) + top-2 + softmax stats
//   Block = 128 threads = 4 waves; each wave owns a 16-token tile (M=16).
//   N = 64 experts = 4 accumulator tiles; K = 2048 (512 WMMA K-steps).
// ---------------------------------------------------------------------------
__global__ __launch_bounds__(128) void moe_router_kernel(
    const float* __restrict__ x,        // (T, C)
    const float* __restrict__ ctx,      // (B, C)
    const float* __restrict__ gate_w,   // (E, C)
    int*   __restrict__ top_idx,        // (T, 2)
    float* __restrict__ top_w,          // (T, 2)
    float* __restrict__ imp_ws,         // (E,) global accumulators (pre-zeroed)
    float* __restrict__ cnt_ws)         // (E,)
{
    __shared__ float s_logits[4][16][65];   // padded stride: no 64-bank conflicts
    __shared__ float s_imp[E_EXP];
    __shared__ float s_cnt[E_EXP];

    const int tid  = threadIdx.x;
    const int wave = tid >> 5;
    const int lane = tid & 31;

    if (tid < E_EXP) { s_imp[tid] = 0.0f; s_cnt[tid] = 0.0f; }

    const int tile_token = blockIdx.x * 64 + wave * 16;   // 64 tokens per block
    const int b          = tile_token >> 12;              // /4096; uniform in tile
    const int half       = lane >> 4;                     // 0: K {0,1}, 1: K {2,3}
    const int l16        = lane & 15;
    const int koff       = half * 2;

    const float* a_row   = x + (size_t)(tile_token + l16) * C_DIM;  // M = l16
    const float* ctx_row = ctx + (size_t)b * C_DIM;
    const float* b_row0  = gate_w + (size_t)(l16 +  0) * C_DIM;     // N = l16 + 16*n
    const float* b_row1  = gate_w + (size_t)(l16 + 16) * C_DIM;
    const float* b_row2  = gate_w + (size_t)(l16 + 32) * C_DIM;
    const float* b_row3  = gate_w + (size_t)(l16 + 48) * C_DIM;

    v8f acc0 = {}, acc1 = {}, acc2 = {}, acc3 = {};

    for (int kc = 0; kc < C_DIM; kc += 32) {
        // stream the next chunk of this wave's x rows (global_prefetch_b8)
        __builtin_prefetch(a_row + (kc + 512 < C_DIM ? kc + 512 : kc), 0, 1);
#pragma unroll
        for (int k0 = kc; k0 < kc + 32; k0 += 4) {
            const int k = k0 + koff;
            // A frag: x + broadcast ctx (16x4 f32 layout)
            v2f a = *(const v2f*)(a_row + k) + *(const v2f*)(ctx_row + k);
            // B frags: gate_w^T columns, K-pair per lane (L2-resident)
            v2f b0 = *(const v2f*)(b_row0 + k);
            v2f b1 = *(const v2f*)(b_row1 + k);
            v2f b2 = *(const v2f*)(b_row2 + k);
            v2f b3 = *(const v2f*)(b_row3 + k);
            acc0 = __builtin_amdgcn_wmma_f32_16x16x4_f32(false, a, false, b0, (short)0, acc0, false, false);
            acc1 = __builtin_amdgcn_wmma_f32_16x16x4_f32(false, a, false, b1, (short)0, acc1, false, false);
            acc2 = __builtin_amdgcn_wmma_f32_16x16x4_f32(false, a, false, b2, (short)0, acc2, false, false);
            acc3 = __builtin_amdgcn_wmma_f32_16x16x4_f32(false, a, false, b3, (short)0, acc3, false, false);
        }
    }

    // C/D layout: VGPR v, lanes 0-15 -> M=v, lanes 16-31 -> M=v+8; N=lane%16
    float* dst = &s_logits[wave][0][0];
#pragma unroll
    for (int v = 0; v < 8; ++v) {
        const int M = v + half * 8;
        dst[M * 65 +  0 + l16] = acc0[v];
        dst[M * 65 + 16 + l16] = acc1[v];
        dst[M * 65 + 32 + l16] = acc2[v];
        dst[M * 65 + 48 + l16] = acc3[v];
    }
    __syncthreads();

    // lanes 0-15 of each wave: one token each
    if (lane < 16) {
        const float* row = &s_logits[wave][lane][0];
        float m1 = -3.4e38f, m2 = -3.4e38f;
        int   i1 = 0, i2 = 0;
#pragma unroll
        for (int e = 0; e < E_EXP; ++e) {
            float v = row[e];
            if (v > m1)      { m2 = m1; i2 = i1; m1 = v; i1 = e; }
            else if (v > m2) { m2 = v;  i2 = e; }
        }
        float sum = 0.0f;
#pragma unroll
        for (int e = 0; e < E_EXP; ++e) sum += __expf(row[e] - m1);
        const float inv = 1.0f / sum;
#pragma unroll
        for (int e = 0; e < E_EXP; ++e)
            atomicAdd(&s_imp[e], __expf(row[e] - m1) * inv);   // ds_add_f32
        atomicAdd(&s_cnt[i1], 1.0f);
        atomicAdd(&s_cnt[i2], 1.0f);

        const int t = tile_token + lane;
        top_idx[2 * t]     = i1;
        top_idx[2 * t + 1] = i2;
        float w1 = 1.0f / (1.0f + __expf(m2 - m1));            // softmax of top-2
        top_w[2 * t]     = w1;
        top_w[2 * t + 1] = 1.0f - w1;
    }
    __syncthreads();

    if (tid < E_EXP) {
        atomicAdd(&imp_ws[tid], s_imp[tid]);                   // global_atomic_add_f32
        atomicAdd(&cnt_ws[tid], s_cnt[tid]);
    }
}

// ---------------------------------------------------------------------------
// Kernel 3: aux_loss = E * sum_e (imp[e]/T) * (cnt[e]/T)
// ---------------------------------------------------------------------------
__global__ __launch_bounds__(64) void aux_loss_kernel(
    const float* __restrict__ imp_ws,
    const float* __restrict__ cnt_ws,
    float* __restrict__ aux_out)
{
    __shared__ float s[E_EXP];
    const int e = threadIdx.x;
    const float invT = 1.0f / (float)T_TOK;
    s[e] = (imp_ws[e] * invT) * (cnt_ws[e] * invT);
    __syncthreads();
    if (e == 0) {
        float sum = 0.0f;
#pragma unroll
        for (int i = 0; i < E_EXP; ++i) sum += s[i];
        *aux_out = (float)E_EXP * sum;
    }
}

// ---------------------------------------------------------------------------
extern "C" void kernel_launch(void* const* d_in, const int* in_sizes, int n_in,
                              void* d_out, int out_size, void* d_ws, size_t ws_size,
                              hipStream_t stream) {
    const float* x      = (const float*)d_in[0];  // (4,4096,2048)
    const float* rc     = (const float*)d_in[1];  // (4,2048)
    const float* gate_w = (const float*)d_in[2];  // (64,2048)
    const float* ctx_w  = (const float*)d_in[3];  // (2048,2048)

    float* ctx_ws = (float*)d_ws;                 // 4*2048 floats
    float* imp_ws = ctx_ws + B_BATCH * C_DIM;     // 64 floats
    float* cnt_ws = imp_ws + E_EXP;               // 64 floats

    int*   idx_out = (int*)d_out;                 // T*2 int32
    float* w_out   = (float*)d_out + 2 * T_TOK;   // T*2 f32
    float* aux_out = w_out + 2 * T_TOK;           // 1 f32

    hipMemsetAsync(imp_ws, 0, 2 * E_EXP * sizeof(float), stream);

    ctx_proj_kernel<<<(B_BATCH * C_DIM) / 256, 256, 0, stream>>>(rc, ctx_w, ctx_ws);
    moe_router_kernel<<<T_TOK / 64, 128, 0, stream>>>(x, ctx_ws, gate_w,
                                                      idx_out, w_out, imp_ws, cnt_ws);
    aux_loss_kernel<<<1, E_EXP, 0, stream>>>(imp_ws, cnt_ws, aux_out);
}